// SortSphericalSignals_83528523972681
// MI455X (gfx1250) — compile-verified
//
#include <hip/hip_runtime.h>

// ---------------------------------------------------------------------------
// SortSphericalSignals: out[z,i,:] = x[z,perm[i],:] where perm comes from the
// block-permutation mixing matrix M (exactly one 1.0 per row).
// Memory-bound: ~453 MB of traffic -> ~19.5us floor at 23.3 TB/s.
// Bulk copy uses the CDNA5 async LDS data-mover path (ASYNCcnt):
//   global_load_async_to_lds_b128 -> s_wait_asynccnt -> global_store_async_from_lds_b128
// Data never passes through VGPRs. Inline asm is used because the clang
// builtins for these ops require LangAS-qualified pointer types that are not
// spellable from HIP source (cuda_device/cuda_shared).
// ---------------------------------------------------------------------------

#define FDIM 864
#define ADIM 4096

#ifndef __has_builtin
#define __has_builtin(x) 0
#endif

#if __has_builtin(__builtin_amdgcn_s_wait_asynccnt)
#define USE_WAIT_BUILTIN 1
#else
#define USE_WAIT_BUILTIN 0
#endif

__device__ __forceinline__ void async_load_b128(const float* gsrc, unsigned lds_off) {
  // VDST = LDS byte address (32-bit VGPR), VADDR = 64-bit global address pair.
  asm volatile("global_load_async_to_lds_b128 %0, %1, off"
               :: "v"(lds_off), "v"(gsrc) : "memory");
}

__device__ __forceinline__ void async_store_b128(float* gdst, unsigned lds_off) {
  // VADDR = 64-bit global address pair, VSRC = LDS byte address (32-bit VGPR).
  asm volatile("global_store_async_from_lds_b128 %0, %1, off"
               :: "v"(gdst), "v"(lds_off) : "memory");
}

__device__ __forceinline__ void wait_async0() {
#if USE_WAIT_BUILTIN
  __builtin_amdgcn_s_wait_asynccnt(0);
#else
  asm volatile("s_wait_asynccnt 0" ::: "memory");
#endif
}

// ---------------------------------------------------------------------------
// Kernel 1: extract permutation from the mixing matrix.
// Exactly one nonzero per row -> exactly one lane writes perm[i].
// ---------------------------------------------------------------------------
__global__ __launch_bounds__(256)
void extract_perm_kernel(const float* __restrict__ M, int* __restrict__ perm) {
  const int i = blockIdx.x;
  const float* row = M + (size_t)i * FDIM;
  for (int j = threadIdx.x; j < FDIM; j += 256) {
    if (row[j] != 0.0f) perm[i] = j;
  }
}

// ---------------------------------------------------------------------------
// Kernel 2: row gather via async memory->LDS->memory DMA path.
// One block per (z,i) output row of 16 KB. Each thread moves 4 x 16B chunks.
// Per-wave LDS regions are disjoint, so only per-wave ASYNCcnt waits needed
// (each wave stores exactly the LDS bytes its own async loads wrote).
// ---------------------------------------------------------------------------
__global__ __launch_bounds__(256)
void gather_rows_async_kernel(const float* __restrict__ x,
                              float* __restrict__ out,
                              const int* __restrict__ perm) {
  __shared__ float tile[ADIM];  // 16 KB

  const int row = blockIdx.x;
  const int z  = row / FDIM;
  const int i  = row - z * FDIM;
  const int j  = perm[i];

  const float* src = x   + ((size_t)z * FDIM + j) * ADIM;
  float*       dst = out + ((size_t)z * FDIM + i) * ADIM;

  const unsigned t = threadIdx.x;
  // Low 32 bits of the generic LDS pointer == byte offset in this WG's LDS.
  const unsigned lds_base = (unsigned)(unsigned long long)(&tile[0]);

#pragma unroll
  for (int k = 0; k < 4; ++k) {
    const unsigned fo = t * 4u + (unsigned)k * 1024u;      // float index
    async_load_b128(src + fo, lds_base + fo * 4u);
  }
  wait_async0();  // this wave's LDS slice now populated

#pragma unroll
  for (int k = 0; k < 4; ++k) {
    const unsigned fo = t * 4u + (unsigned)k * 1024u;
    async_store_b128(dst + fo, lds_base + fo * 4u);
  }
  wait_async0();  // drain before ENDPGM (implicit wait-idle also covers this)
}

// ---------------------------------------------------------------------------
extern "C" void kernel_launch(void* const* d_in, const int* in_sizes, int n_in,
                              void* d_out, int out_size, void* d_ws, size_t ws_size,
                              hipStream_t stream) {
  const float* x = (const float*)d_in[0];
  const float* M = (const float*)d_in[1];
  float* out = (float*)d_out;
  int* perm = (int*)d_ws;  // 864 ints of scratch

  // 1) Recover permutation from M (deterministic, recomputed every call).
  extract_perm_kernel<<<FDIM, 256, 0, stream>>>(M, perm);

  // 2) Bulk gather-copy at HBM rate via async LDS path.
  const int B = in_sizes[0] / (FDIM * ADIM);  // 16
  gather_rows_async_kernel<<<B * FDIM, 256, 0, stream>>>(x, out, perm);
}